// ATSSMatcher_51470888075345
// MI455X (gfx1250) — compile-verified
//
#include <hip/hip_runtime.h>
#include <hip/hip_bf16.h>
#include <float.h>
#include <math.h>

// ---------------------------------------------------------------------------
// ATSS matcher for MI455X (gfx1250).
// Dense part (rank keys |a|^2 - 2*a.g over 299520x64) via V_WMMA_F32_16X16X4_F32.
// IoU evaluated only at <=2304 candidate pairs; conflict resolution via u64 atomicMax.
// ---------------------------------------------------------------------------

typedef float v2f __attribute__((ext_vector_type(2)));
typedef float v8f __attribute__((ext_vector_type(8)));

#define NUM_GT     64
#define K_TOTAL    36          // 9 candidates x 4 levels
#define CHUNK      2048        // anchors per phase-A block
#define NCHUNKS    147         // 128 + 16 + 2 + 1
#define GTILES     4           // 64 GTs / 16 per WMMA tile
#define BLOCK_A    256
#define WAVES_A    8
#define MAXENT     1152        // 128 chunks * 9 entries (level 0)

// Static FPN level table: sizes {262144, 32768, 4096, 512}
// starts {0, 262144, 294912, 299008}; chunk-id starts {0, 128, 144, 146}

__device__ __forceinline__ unsigned long long pack_key(float iou, int g) {
    // iou >= 0 so its bit pattern is monotone as u32; low word makes earlier
    // GT indices win ties (argmax tie -> lowest index) and keeps key nonzero.
    return ((unsigned long long)__float_as_uint(iou) << 32) |
           (unsigned long long)(unsigned)(NUM_GT - g);
}

// ---------------------------------------------------------------------------
// Kernel 0a: zero the per-anchor argmax keys (harness does not re-zero ws).
// ---------------------------------------------------------------------------
__global__ void atss_init(unsigned long long* __restrict__ akey, int n) {
    int i = blockIdx.x * blockDim.x + threadIdx.x;
    if (i < n) akey[i] = 0ULL;
}

// ---------------------------------------------------------------------------
// Kernel 0b: precompute (cx, cy, cz, |c|^2) per anchor once; kernel A then
// needs only one b128 load per anchor per GT-tile pass.
// ---------------------------------------------------------------------------
__global__ void atss_centers(const float* __restrict__ anchors,
                             float4* __restrict__ ctr, int n) {
    const int i = blockIdx.x * blockDim.x + threadIdx.x;
    if (i >= n) return;
    const float* ab = anchors + (size_t)i * 6;
    const float cx = 0.5f * (ab[0] + ab[2]);
    const float cy = 0.5f * (ab[1] + ab[3]);
    const float cz = 0.5f * (ab[4] + ab[5]);
    ctr[i] = make_float4(cx, cy, cz, cx * cx + cy * cy + cz * cz);
}

// ---------------------------------------------------------------------------
// Kernel A: per (chunk, gt-tile) block. WMMA f32 16x16x4 computes a 16-anchor
// x 16-GT tile of rank keys; each lane keeps an UNSORTED top-9 (tracked max)
// for its GT column; lists are merged across the block in LDS.
// A-fragment loads are software-pipelined (double-buffered) so the top-9
// insertion work overlaps the next tile's L2 load latency.
// ---------------------------------------------------------------------------
__global__ void __launch_bounds__(BLOCK_A)
atss_topk_chunks(const float4* __restrict__ ctr,
                 const float* __restrict__ gts,
                 float* __restrict__ okey,
                 int*   __restrict__ oidx) {
    const int chunkId = blockIdx.x;
    const int gtile   = blockIdx.y;
    const int tid     = threadIdx.x;
    const int lane    = tid & 31;
    const int wave    = tid >> 5;
    const int half    = lane >> 4;   // 0: K0/K1 lanes, 1: K2/K3 lanes
    const int l16     = lane & 15;

    // chunk -> (level, anchor range)
    int lvl;
    if      (chunkId >= 146) lvl = 3;
    else if (chunkId >= 144) lvl = 2;
    else if (chunkId >= 128) lvl = 1;
    else                     lvl = 0;
    const int lsize[4]  = {262144, 32768, 4096, 512};
    const int lstart[4] = {0, 262144, 294912, 299008};
    const int cstart[4] = {0, 128, 144, 146};
    const int offInLvl = (chunkId - cstart[lvl]) * CHUNK;
    const int aStart   = lstart[lvl] + offInLvl;
    const int aCount   = (lsize[lvl] - offInLvl < CHUNK) ? (lsize[lvl] - offInLvl) : CHUNK;
    const int nTiles   = aCount >> 4;   // all level sizes are multiples of 16

    // B fragment: column n = GT, B[:,n] = (-2gx, -2gy, -2gz, 1)
    const int gcol = gtile * 16 + l16;
    const float* gb = gts + (size_t)gcol * 6;
    const float gx = 0.5f * (gb[0] + gb[2]);
    const float gy = 0.5f * (gb[1] + gb[3]);
    const float gz = 0.5f * (gb[4] + gb[5]);
    v2f bfrag;
    bfrag.x = half ? (-2.0f * gz) : (-2.0f * gx);
    bfrag.y = half ? 1.0f         : (-2.0f * gy);

    // per-lane UNSORTED top-9 of (key, anchor) for GT column l16; track max.
    float kl[9]; int il[9];
#pragma unroll
    for (int i = 0; i < 9; ++i) { kl[i] = FLT_MAX; il[i] = -1; }
    float kmax = FLT_MAX;
    int   kpos = 0;

    // software-pipelined A-fragment stream (wave-uniform guards keep EXEC full)
    float4 c4 = make_float4(0.f, 0.f, 0.f, 0.f);
    if (wave < nTiles)
        c4 = ctr[(size_t)(aStart + wave * 16 + l16)];

    for (int tile = wave; tile < nTiles; tile += WAVES_A) {
        const int aBase = aStart + tile * 16;
        const float4 cur = c4;
        const int nextTile = tile + WAVES_A;
        if (nextTile < nTiles) {
            // issue next tile's load early; also prefetch one further ahead
            c4 = ctr[(size_t)(aStart + nextTile * 16 + l16)];
            __builtin_prefetch(ctr + (size_t)(aStart + (nextTile + WAVES_A) * 16 + l16), 0, 1);
        }

        // A fragment: row m = anchor, A[m,:] = (cx, cy, cz, |c|^2)
        v2f afrag;
        afrag.x = half ? cur.z : cur.x;
        afrag.y = half ? cur.w : cur.y;

        v8f c = {0.f, 0.f, 0.f, 0.f, 0.f, 0.f, 0.f, 0.f};
        c = __builtin_amdgcn_wmma_f32_16x16x4_f32(
                /*neg_a=*/false, afrag, /*neg_b=*/false, bfrag,
                /*c_mod=*/(short)0, c, /*reuse_a=*/false, /*reuse_b=*/false);

        // C layout: VGPR r -> row (half*8 + r), column l16
#pragma unroll
        for (int r = 0; r < 8; ++r) {
            const float key  = c[r];
            const int   aidx = aBase + half * 8 + r;
            if (key < kmax) {
                // replace current-max slot (cndmask chain, no array rotation)
#pragma unroll
                for (int q = 0; q < 9; ++q)
                    if (q == kpos) { kl[q] = key; il[q] = aidx; }
                // recompute tracked max
                float m = kl[0]; int mp = 0;
#pragma unroll
                for (int q = 1; q < 9; ++q)
                    if (kl[q] > m) { m = kl[q]; mp = q; }
                kmax = m; kpos = mp;
            }
        }
    }

    // Block-level merge: list id = tid; lists for GT g live at ids g + 16*j.
    __shared__ float lkey[BLOCK_A * 9];
    __shared__ int   lidx[BLOCK_A * 9];
#pragma unroll
    for (int s = 0; s < 9; ++s) { lkey[tid * 9 + s] = kl[s]; lidx[tid * 9 + s] = il[s]; }
    __syncthreads();

    if (tid < 16) {
        const int g = tid;
        const size_t obase = ((size_t)chunkId * NUM_GT + (size_t)gtile * 16 + g) * 9;
        // 9-round selection over the 144 (unsorted) entries for this GT
        for (int s = 0; s < 9; ++s) {
            float best = FLT_MAX; int bp = -1;
            for (int j = 0; j < 16; ++j) {
                const int base = (g + 16 * j) * 9;
                for (int k = 0; k < 9; ++k) {
                    const float v = lkey[base + k];
                    if (v < best) { best = v; bp = base + k; }
                }
            }
            float ok = FLT_MAX; int oi = -1;
            if (bp >= 0) { ok = best; oi = lidx[bp]; lkey[bp] = FLT_MAX; }
            okey[obase + s] = ok;
            oidx[obase + s] = oi;
        }
    }
}

// ---------------------------------------------------------------------------
// Kernel B: one block per GT. Merge chunk top-9s per level -> 36 candidates,
// compute IoU, adaptive threshold, positivity, u64 atomicMax assignment.
// ---------------------------------------------------------------------------
__global__ void __launch_bounds__(256)
atss_select_assign(const float* __restrict__ anchors,
                   const float* __restrict__ gts,
                   const float* __restrict__ okey,
                   const int*   __restrict__ oidx,
                   unsigned long long* __restrict__ akey) {
    const int g   = blockIdx.x;
    const int tid = threadIdx.x;

    __shared__ float ek[MAXENT];
    __shared__ int   ei[MAXENT];
    __shared__ float rk[256];
    __shared__ int   rp[256];
    __shared__ int   candIdx[K_TOTAL];
    __shared__ float candIou[K_TOTAL];
    __shared__ int   candIn[K_TOTAL];
    __shared__ float G[6];
    __shared__ float sThr;

    const int cstart[4] = {0, 128, 144, 146};
    const int ccount[4] = {128, 16, 2, 1};

    for (int lvl = 0; lvl < 4; ++lvl) {
        const int nent = ccount[lvl] * 9;
        for (int e = tid; e < nent; e += 256) {
            const int c = cstart[lvl] + e / 9;
            const int s = e % 9;
            const size_t q = ((size_t)c * NUM_GT + g) * 9 + s;
            const int ai = oidx[q];
            ek[e] = (ai >= 0) ? okey[q] : FLT_MAX;
            ei[e] = ai;
        }
        __syncthreads();

        for (int s = 0; s < 9; ++s) {
            // parallel argmin over nent entries (tie -> smallest entry pos)
            float bk = FLT_MAX; int bp = -1;
            for (int e = tid; e < nent; e += 256) {
                const float v = ek[e];
                if (v < bk) { bk = v; bp = e; }
            }
            rk[tid] = bk; rp[tid] = bp;
            __syncthreads();
            for (int off = 128; off > 0; off >>= 1) {
                if (tid < off) {
                    const float v = rk[tid + off]; const int p = rp[tid + off];
                    if (v < rk[tid] ||
                        (v == rk[tid] && p >= 0 && (rp[tid] < 0 || p < rp[tid]))) {
                        rk[tid] = v; rp[tid] = p;
                    }
                }
                __syncthreads();
            }
            if (tid == 0) {
                const int p = rp[0];
                const int ai = (p >= 0) ? ei[p] : -1;
                candIdx[lvl * 9 + s] = (ai >= 0) ? ai : 0;
                if (p >= 0) ek[p] = FLT_MAX;   // remove for next round
            }
            __syncthreads();
        }
        __syncthreads();
    }

    if (tid < 6) G[tid] = gts[(size_t)g * 6 + tid];
    __syncthreads();

    if (tid < K_TOTAL) {
        const int a = candIdx[tid];
        const float* ab = anchors + (size_t)a * 6;
        const float ax1 = ab[0], ay1 = ab[1], ax2 = ab[2];
        const float ay2 = ab[3], az1 = ab[4], az2 = ab[5];
        const float gx1 = G[0], gy1 = G[1], gx2 = G[2];
        const float gy2 = G[3], gz1 = G[4], gz2 = G[5];
        const float v1 = (ax2 - ax1) * (ay2 - ay1) * (az2 - az1);
        const float v2 = (gx2 - gx1) * (gy2 - gy1) * (gz2 - gz1);
        float wx = fminf(ax2, gx2) - fmaxf(ax1, gx1); wx = fmaxf(wx, 0.0f);
        float wy = fminf(ay2, gy2) - fmaxf(ay1, gy1); wy = fmaxf(wy, 0.0f);
        float wz = fminf(az2, gz2) - fmaxf(az1, gz1); wz = fmaxf(wz, 0.0f);
        const float inter = wx * wy * wz;
        candIou[tid] = inter / (v1 + v2 - inter + 1e-6f);
        const float cx = 0.5f * (ax1 + ax2);
        const float cy = 0.5f * (ay1 + ay2);
        const float cz = 0.5f * (az1 + az2);
        candIn[tid] = (cx >= gx1) & (cx <= gx2) &
                      (cy >= gy1) & (cy <= gy2) &
                      (cz >= gz1) & (cz <= gz2);
    }
    __syncthreads();

    if (tid == 0) {
        float s = 0.0f;
        for (int i = 0; i < K_TOTAL; ++i) s += candIou[i];
        const float mean = s / (float)K_TOTAL;
        float v = 0.0f;
        for (int i = 0; i < K_TOTAL; ++i) {
            const float d = candIou[i] - mean;
            v += d * d;
        }
        v /= (float)(K_TOTAL - 1);   // unbiased (ddof=1)
        sThr = fmaxf(mean + sqrtf(v), 0.0f);   // MIN_IOU = 0
    }
    __syncthreads();

    if (tid < K_TOTAL) {
        if (candIn[tid] && candIou[tid] >= sThr) {
            atomicMax(akey + candIdx[tid], pack_key(candIou[tid], g));
        }
    }
}

// ---------------------------------------------------------------------------
// Kernel C: decode per-anchor keys -> (matched_gt, matched_iou, label).
// ---------------------------------------------------------------------------
__global__ void atss_finalize(const unsigned long long* __restrict__ akey,
                              int* __restrict__ outIdx,
                              float* __restrict__ outIou,
                              int* __restrict__ outLab, int n) {
    const int i = blockIdx.x * blockDim.x + threadIdx.x;
    if (i >= n) return;
    const unsigned long long k = akey[i];
    if (k == 0ULL) {
        outIdx[i] = -1; outIou[i] = 0.0f; outLab[i] = 0;
    } else {
        outIdx[i] = NUM_GT - (int)(unsigned)(k & 0xFFFFFFFFULL);
        outIou[i] = __uint_as_float((unsigned)(k >> 32));
        outLab[i] = 1;
    }
}

// ---------------------------------------------------------------------------
extern "C" void kernel_launch(void* const* d_in, const int* in_sizes, int n_in,
                              void* d_out, int out_size, void* d_ws, size_t ws_size,
                              hipStream_t stream) {
    const float* gts     = (const float*)d_in[0];   // gt_boxes (64, 6)
    const float* anchors = (const float*)d_in[1];   // anchors  (N, 6)
    const int N = in_sizes[1] / 6;                  // 299520

    // workspace layout (16B-aligned base):
    //   [float4 ctr (N)] [u64 akey (N)] [f32 okey (147*64*9)] [i32 oidx (147*64*9)]
    float4* ctrBuf = (float4*)d_ws;
    unsigned long long* akey = (unsigned long long*)(ctrBuf + N);
    float* okeyBuf = (float*)(akey + N);
    int*   oidxBuf = (int*)(okeyBuf + (size_t)NCHUNKS * NUM_GT * 9);

    // outputs: tuple concat (matched_gt_indices:i32, matched_iou:f32, labels:i32)
    int*   outIdx = (int*)d_out;
    float* outIou = (float*)d_out + N;
    int*   outLab = (int*)d_out + 2 * (size_t)N;

    atss_init<<<(N + 255) / 256, 256, 0, stream>>>(akey, N);
    atss_centers<<<(N + 255) / 256, 256, 0, stream>>>(anchors, ctrBuf, N);
    dim3 gA(NCHUNKS, GTILES);
    atss_topk_chunks<<<gA, BLOCK_A, 0, stream>>>(ctrBuf, gts, okeyBuf, oidxBuf);
    atss_select_assign<<<NUM_GT, 256, 0, stream>>>(anchors, gts, okeyBuf, oidxBuf, akey);
    atss_finalize<<<(N + 255) / 256, 256, 0, stream>>>(akey, outIdx, outIou, outLab, N);
}